// AttentionUnit_1924145348908
// MI455X (gfx1250) — compile-verified
//
#include <hip/hip_runtime.h>
#include <math.h>

typedef __attribute__((ext_vector_type(2))) float v2f;
typedef __attribute__((ext_vector_type(8))) float v8f;
typedef __attribute__((ext_vector_type(4))) unsigned int u32x4;
typedef __attribute__((ext_vector_type(8))) int i32x8;
typedef __attribute__((ext_vector_type(4))) int i32x4;

#define BATCH 8192
#define LHIST 200
#define DIM   128
#define SKSTR 132   // LDS row stride (floats) = 128 data + 4 pad (TDM pad fields)

// ---------------------------------------------------------------------------
// Kernel A: one wave (32 threads) per block; block handles 16 batch rows.
//   q' = query @ Wq^T + bq          (16x128 stripe, f32 WMMA 16x16x4)
//   qk = q' @ Wk                    (16x128 stripe, f32 WMMA 16x16x4)
// bk is dropped: its score contribution is constant over L -> cancels in
// the softmax, so neither output depends on it.
// ---------------------------------------------------------------------------
__global__ __launch_bounds__(32)
void proj_kernel(const float* __restrict__ query,
                 const float* __restrict__ Wq,
                 const float* __restrict__ bq,
                 const float* __restrict__ Wk,
                 float* __restrict__ qk)
{
    __shared__ float qp[16 * DIM];   // q' stripe

    const int lane = threadIdx.x;
    const int m    = lane & 15;      // M index (A/C) == N index (B/C)
    const int half = lane >> 4;      // 0 or 1
    const int koff = half * 2;       // K sub-offset within a 4-wide K step
    const int row0 = blockIdx.x * 16;

    // ---- GEMM1: q' = query @ Wq^T  (+ bq) ----
    v8f acc[8] = {};
    for (int kk = 0; kk < 32; ++kk) {
        const int kb = kk * 4;
        v2f a = *(const v2f*)(query + (size_t)(row0 + m) * DIM + kb + koff);
        #pragma unroll
        for (int j = 0; j < 8; ++j) {
            v2f b = *(const v2f*)(Wq + (size_t)(j * 16 + m) * DIM + kb + koff);
            acc[j] = __builtin_amdgcn_wmma_f32_16x16x4_f32(
                false, a, false, b, (short)0, acc[j], false, false);
        }
    }
    #pragma unroll
    for (int j = 0; j < 8; ++j) {
        float bias = bq[j * 16 + m];
        #pragma unroll
        for (int v = 0; v < 8; ++v)
            qp[(v + half * 8) * DIM + j * 16 + m] = acc[j][v] + bias;
    }
    __syncthreads();

    // ---- GEMM2: qk = q' @ Wk ----
    v8f acc2[8] = {};
    for (int kk = 0; kk < 32; ++kk) {
        const int kb = kk * 4;
        v2f a = *(const v2f*)(qp + m * DIM + kb + koff);
        #pragma unroll
        for (int j = 0; j < 8; ++j) {
            v2f b;
            b[0] = Wk[(size_t)(kb + koff + 0) * DIM + j * 16 + m];
            b[1] = Wk[(size_t)(kb + koff + 1) * DIM + j * 16 + m];
            acc2[j] = __builtin_amdgcn_wmma_f32_16x16x4_f32(
                false, a, false, b, (short)0, acc2[j], false, false);
        }
    }
    #pragma unroll
    for (int j = 0; j < 8; ++j)
        #pragma unroll
        for (int v = 0; v < 8; ++v)
            qk[(size_t)(row0 + v + half * 8) * DIM + j * 16 + m] = acc2[j][v];
}

// ---------------------------------------------------------------------------
// Kernel B: one block per batch row b. TDM-DMA keys[b] (200x128 f32) into
// padded LDS (hardware row padding -> stride 132), compute scores against
// qk[b], softmax over L, weighted sum of the original keys.
// ---------------------------------------------------------------------------
__global__ __launch_bounds__(256)
void attn_kernel(const float* __restrict__ keys,
                 const float* __restrict__ qk,
                 float* __restrict__ out_wsum,
                 float* __restrict__ out_attn)
{
    extern __shared__ float smem[];
    float* sk  = smem;                  // LHIST * SKSTR (TDM destination)
    float* sqk = sk + LHIST * SKSTR;    // DIM
    float* ssc = sqk + DIM;             // 256 (200 used)
    float* red = ssc + 256;             // 256

    const int t = threadIdx.x;
    const int b = blockIdx.x;

    // ---- TDM: DMA keys[b] global -> LDS with row padding (wave 0 only) ----
    if (t < 32) {
        const unsigned long long gaddr =
            (unsigned long long)(const void*)(keys + (size_t)b * (LHIST * DIM));
        const unsigned int lds_off =
            (unsigned int)(unsigned long long)(const void*)sk;  // LDS VA[31:0] = byte offset

        u32x4 g0;
        g0[0] = 1u;                                   // count=1, user mode, no gather
        g0[1] = lds_off;                              // lds_addr
        g0[2] = (unsigned int)(gaddr & 0xFFFFFFFFull);          // global_addr[31:0]
        g0[3] = (unsigned int)((gaddr >> 32) & 0x01FFFFFFull)   // global_addr[56:32]
                | (2u << 30);                         // type = 2 ("image")

        i32x8 g1;
        g1[0] = (2 << 16)      // data_size = 4 bytes
              | (1 << 20)      // pad_enable
              | (6 << 22)      // pad_interval: 128 DWORDs per row
              | (3 << 25);     // pad_amount: 4 DWORDs -> LDS stride 132 floats
        g1[1] = (DIM   << 16); // tensor_dim0[15:0] in bits [63:48]
        g1[2] = (LHIST << 16); // tensor_dim0[31:16]=0, tensor_dim1[15:0]=200
        g1[3] = (DIM   << 16); // tensor_dim1[31:16]=0, tile_dim0=128
        g1[4] = LHIST;         // tile_dim1=200, tile_dim2=0
        g1[5] = DIM;           // tensor_dim0_stride[31:0] = 128
        g1[6] = 0;             // stride[47:32]=0, tensor_dim1_stride lo = 0
        g1[7] = 0;

        i32x4 g2 = {};         // 2-D tensor: groups 2/3 unused
        i32x4 g3 = {};
        i32x8 g4 = {};         // extra operand of 6-arg builtin: unused here

        __builtin_amdgcn_tensor_load_to_lds(g0, g1, g2, g3, g4, 0);
        __builtin_amdgcn_s_wait_tensorcnt(0);
    }
    if (t < DIM) sqk[t] = qk[(size_t)b * DIM + t];
    __syncthreads();

    // scores: ssc[l] = <keys[b,l], qk[b]>
    if (t < LHIST) {
        const float* row = sk + t * SKSTR;
        float acc = 0.f;
        #pragma unroll 4
        for (int j = 0; j < DIM; ++j) acc += row[j] * sqk[j];
        ssc[t] = acc;
    }
    __syncthreads();

    // softmax over L: max-reduce
    float v = (t < LHIST) ? ssc[t] : -INFINITY;
    red[t] = v;
    __syncthreads();
    for (int s = 128; s > 0; s >>= 1) {
        if (t < s) red[t] = fmaxf(red[t], red[t + s]);
        __syncthreads();
    }
    const float smax = red[0];
    __syncthreads();

    // exp + sum-reduce
    const float e = (t < LHIST) ? __expf(ssc[t] - smax) : 0.f;
    red[t] = e;
    __syncthreads();
    for (int s = 128; s > 0; s >>= 1) {
        if (t < s) red[t] += red[t + s];
        __syncthreads();
    }
    const float inv = 1.0f / red[0];
    __syncthreads();

    if (t < LHIST) {
        const float a = e * inv;
        ssc[t] = a;
        out_attn[(size_t)b * LHIST + t] = a;
    }
    __syncthreads();

    // weighted sum over the ORIGINAL keys: out[d] = sum_l attn[l]*keys[b,l,d]
    const int d = t & 127;
    const int h = t >> 7;
    float acc = 0.f;
    for (int l = h * 100; l < h * 100 + 100; ++l)
        acc += ssc[l] * sk[l * SKSTR + d];
    red[t] = acc;
    __syncthreads();
    if (t < 128) out_wsum[(size_t)b * DIM + t] = red[t] + red[t + 128];
}

// ---------------------------------------------------------------------------
extern "C" void kernel_launch(void* const* d_in, const int* in_sizes, int n_in,
                              void* d_out, int out_size, void* d_ws, size_t ws_size,
                              hipStream_t stream)
{
    const float* query = (const float*)d_in[0];
    const float* keys  = (const float*)d_in[1];
    const float* Wq    = (const float*)d_in[2];
    const float* bq    = (const float*)d_in[3];
    const float* Wk    = (const float*)d_in[4];
    // d_in[5] (bk) intentionally unused: constant shift cancels in softmax.

    float* qk = (float*)d_ws;                      // B*D floats = 4 MB scratch
    float* out_wsum = (float*)d_out;               // [B, D]
    float* out_attn = (float*)d_out + (size_t)BATCH * DIM;  // [B, L]

    proj_kernel<<<BATCH / 16, 32, 0, stream>>>(query, Wq, bq, Wk, qk);

    const size_t shmem = (size_t)(LHIST * SKSTR + DIM + 256 + 256) * sizeof(float);
    attn_kernel<<<BATCH, 256, shmem, stream>>>(keys, qk, out_wsum, out_attn);
}